// JKEGINEmbeddingNet_4612794876598
// MI455X (gfx1250) — compile-verified
//
#include <hip/hip_runtime.h>
#include <hip/hip_bf16.h>
#include <stdint.h>

#define N_NODES   50000
#define N_EDGES   640000
#define NUM_FEAT  32
#define DIM       128
#define NUM_LAYERS 5
#define NUM_GRAPHS 64
#define BN_EPS    1e-5f
#define NEG_SLOPE 0.01f
#define PROC_STEPS 5

typedef float v2f __attribute__((ext_vector_type(2)));
typedef float v8f __attribute__((ext_vector_type(8)));

__device__ __forceinline__ float leakyf(float v) {
    return v >= 0.0f ? v : NEG_SLOPE * v;
}
__device__ __forceinline__ float sigmoidf(float v) {
    return 1.0f / (1.0f + expf(-v));
}

// ---------------------------------------------------------------- utilities
__global__ void fill_zero_kernel(float* p, int n) {
    int i = blockIdx.x * blockDim.x + threadIdx.x;
    if (i < n) p[i] = 0.0f;
}

// Pad 32-feature input into 128-wide feature buffer (zeros in cols 32..127)
__global__ void pad_x_kernel(const float* __restrict__ xin, float* __restrict__ xbuf) {
    int i = blockIdx.x * blockDim.x + threadIdx.x;
    if (i >= N_NODES * DIM) return;
    int n = i >> 7, d = i & 127;
    xbuf[i] = (d < NUM_FEAT) ? xin[n * NUM_FEAT + d] : 0.0f;
}

// Pad 32x128 weight into 128x128 (zero rows 32..127). Layout identical flat.
__global__ void pad_w_kernel(const float* __restrict__ w, float* __restrict__ wpad) {
    int i = blockIdx.x * blockDim.x + threadIdx.x;
    if (i >= DIM * DIM) return;
    wpad[i] = (i < NUM_FEAT * DIM) ? w[i] : 0.0f;
}

// Re-swizzle a 128x128 row-major weight (K x N) into per-lane WMMA B-fragment
// order: Wf[((nt*32 + k4)*32 + lane)*2 + v] = W[(k4*4 + 2*(lane>>4) + v)*128 + nt*16 + (lane&15)]
// so each wave's B fragment for (tile nt, k-step k4) is one contiguous 256B b64 load.
__global__ void wfrag_kernel(const float* __restrict__ W, float* __restrict__ Wf) {
    int i = blockIdx.x * blockDim.x + threadIdx.x;
    if (i >= DIM * DIM) return;
    int v    = i & 1;
    int lane = (i >> 1) & 31;
    int k4   = (i >> 6) & 31;
    int nt   = i >> 11;
    int k = k4 * 4 + 2 * (lane >> 4) + v;
    int n = nt * 16 + (lane & 15);
    Wf[i] = W[k * DIM + n];
}

// h_pre = (1+eps_i) * x
__global__ void scale_x_kernel(const float* __restrict__ x, const float* __restrict__ eps_p,
                               float* __restrict__ h, int total) {
    int i = blockIdx.x * blockDim.x + threadIdx.x;
    if (i >= total) return;
    h[i] = (1.0f + eps_p[0]) * x[i];
}

// Scatter-add x[src] into h[dst]; one wave per edge, 4 floats per lane.
__global__ void edge_scatter_kernel(const float* __restrict__ x,
                                    const int* __restrict__ src,
                                    const int* __restrict__ dst,
                                    float* __restrict__ h) {
    int gid = blockIdx.x * blockDim.x + threadIdx.x;
    int e = gid >> 5;
    if (e >= N_EDGES) return;
    int lane = gid & 31;
    int s = src[e], d = dst[e];
    const float* xr = x + (size_t)s * DIM + lane * 4;
    float*       hr = h + (size_t)d * DIM + lane * 4;
    atomicAdd(hr + 0, xr[0]);
    atomicAdd(hr + 1, xr[1]);
    atomicAdd(hr + 2, xr[2]);
    atomicAdd(hr + 3, xr[3]);
}

// ---------------------------------------------------------------- WMMA GEMM
// out[rows x 128] = leaky(A[rows x 128] @ W[128 x 128] + bias), f32 WMMA 16x16x4.
// 4 waves / block; each wave owns a 16-row x 64-col strip = 4 accumulator tiles.
// One A-fragment load feeds 4 WMMAs; B fragments come pre-swizzled (Wf) as
// single coalesced b64 loads. rows % 16 == 0.
__global__ __launch_bounds__(128) void wmma_gemm4_leaky_kernel(
    const float* __restrict__ A, const float* __restrict__ Wf,
    const float* __restrict__ bias, float* __restrict__ out, int rows)
{
    int wave = threadIdx.x >> 5;
    int lane = threadIdx.x & 31;
    int m0 = (blockIdx.x * 4 + wave) * 16;
    if (m0 >= rows) return;           // wave-uniform: EXEC stays all-ones
    int lm = lane & 15;
    int lh = lane >> 4;
    int tb = blockIdx.y * 4;          // first of 4 n-tiles handled by this wave

    v8f acc0 = {}, acc1 = {}, acc2 = {}, acc3 = {};
    const float* arow = A + (size_t)(m0 + lm) * DIM;
    const float* wf   = Wf + (size_t)tb * 32 * 64 + lane * 2;

    #pragma unroll
    for (int k4 = 0; k4 < 32; ++k4) {
        v2f a;
        a.x = arow[k4 * 4 + 2 * lh];
        a.y = arow[k4 * 4 + 2 * lh + 1];
        v2f b0, b1, b2, b3;
        b0.x = wf[(0 * 32 + k4) * 64 + 0]; b0.y = wf[(0 * 32 + k4) * 64 + 1];
        b1.x = wf[(1 * 32 + k4) * 64 + 0]; b1.y = wf[(1 * 32 + k4) * 64 + 1];
        b2.x = wf[(2 * 32 + k4) * 64 + 0]; b2.y = wf[(2 * 32 + k4) * 64 + 1];
        b3.x = wf[(3 * 32 + k4) * 64 + 0]; b3.y = wf[(3 * 32 + k4) * 64 + 1];
        acc0 = __builtin_amdgcn_wmma_f32_16x16x4_f32(false, a, false, b0, (short)0, acc0, false, false);
        acc1 = __builtin_amdgcn_wmma_f32_16x16x4_f32(false, a, false, b1, (short)0, acc1, false, false);
        acc2 = __builtin_amdgcn_wmma_f32_16x16x4_f32(false, a, false, b2, (short)0, acc2, false, false);
        acc3 = __builtin_amdgcn_wmma_f32_16x16x4_f32(false, a, false, b3, (short)0, acc3, false, false);
    }

    #pragma unroll
    for (int t = 0; t < 4; ++t) {
        v8f acc = (t == 0) ? acc0 : (t == 1) ? acc1 : (t == 2) ? acc2 : acc3;
        int n0 = (tb + t) * 16;
        float bn = bias[n0 + lm];
        #pragma unroll
        for (int v = 0; v < 8; ++v) {
            int m = m0 + v + 8 * lh;
            out[(size_t)m * DIM + n0 + lm] = leakyf(acc[v] + bn);
        }
    }
}

// ---------------------------------------------------------------- BatchNorm
__global__ __launch_bounds__(128) void bn_stats_kernel(const float* __restrict__ h,
                                                       double* __restrict__ sum,
                                                       double* __restrict__ sumsq) {
    int d = threadIdx.x;
    double s = 0.0, s2 = 0.0;
    for (int r = blockIdx.x; r < N_NODES; r += gridDim.x) {
        float v = h[(size_t)r * DIM + d];
        s += (double)v;
        s2 += (double)v * (double)v;
    }
    atomicAdd(&sum[d], s);
    atomicAdd(&sumsq[d], s2);
}

__global__ __launch_bounds__(128) void bn_apply_kernel(const double* __restrict__ sum,
                                                       const double* __restrict__ sumsq,
                                                       const float* __restrict__ gamma,
                                                       const float* __restrict__ beta,
                                                       float* __restrict__ scale,
                                                       float* __restrict__ shift) {
    int d = threadIdx.x;
    double mu = sum[d] / (double)N_NODES;
    double var = sumsq[d] / (double)N_NODES - mu * mu;
    float sc = gamma[d] * rsqrtf((float)var + BN_EPS);
    scale[d] = sc;
    shift[d] = beta[d] - (float)mu * sc;
}

__global__ void bn_norm_kernel(const float* __restrict__ h, const float* __restrict__ scale,
                               const float* __restrict__ shift, float* __restrict__ xout,
                               float* __restrict__ xmax, int first, int total) {
    int i = blockIdx.x * blockDim.x + threadIdx.x;
    if (i >= total) return;
    int d = i & 127;
    float v = h[i] * scale[d] + shift[d];
    xout[i] = v;
    xmax[i] = first ? v : fmaxf(xmax[i], v);
}

// ---------------------------------------------------------------- Set2Set
// Per-graph node ranges from sorted batch vector.
__global__ void bounds_kernel(const int* __restrict__ batch, int* __restrict__ startv,
                              int* __restrict__ endv) {
    int i = blockIdx.x * blockDim.x + threadIdx.x;
    if (i >= N_NODES) return;
    int b = batch[i];
    if (i == 0) startv[b] = 0;
    else {
        int pb = batch[i - 1];
        if (pb != b) { endv[pb] = i; startv[b] = i; }
    }
    if (i == N_NODES - 1) endv[b] = N_NODES;
}

// One block per graph; thread j owns one hidden unit, computes all 4 gates.
__global__ __launch_bounds__(128) void lstm_kernel(const float* __restrict__ inp, int in_dim,
                                                   const float* __restrict__ Wih,
                                                   const float* __restrict__ Whh,
                                                   const float* __restrict__ bih,
                                                   const float* __restrict__ bhh,
                                                   float* __restrict__ h, float* __restrict__ c) {
    int b = blockIdx.x, j = threadIdx.x;
    __shared__ float s_in[2 * DIM];
    __shared__ float s_h[DIM];
    for (int k = j; k < in_dim; k += DIM) s_in[k] = inp[(size_t)b * in_dim + k];
    s_h[j] = h[(size_t)b * DIM + j];
    __syncthreads();
    float g4[4];
    #pragma unroll
    for (int gi = 0; gi < 4; ++gi) {
        int row = gi * DIM + j;
        float acc = bih[row] + bhh[row];
        const float* wi = Wih + (size_t)row * in_dim;
        for (int k = 0; k < in_dim; ++k) acc += s_in[k] * wi[k];
        const float* wh = Whh + (size_t)row * DIM;
        for (int k = 0; k < DIM; ++k) acc += s_h[k] * wh[k];
        g4[gi] = acc;
    }
    float ig = sigmoidf(g4[0]);
    float fg = sigmoidf(g4[1]);
    float gg = tanhf(g4[2]);
    float og = sigmoidf(g4[3]);
    float cn = fg * c[(size_t)b * DIM + j] + ig * gg;
    float hn = og * tanhf(cn);
    c[(size_t)b * DIM + j] = cn;
    h[(size_t)b * DIM + j] = hn;
}

// e[n] = dot(x[n], q[batch[n]]); one wave per node.
__global__ void escore_kernel(const float* __restrict__ x, const float* __restrict__ q,
                              const int* __restrict__ batch, float* __restrict__ e) {
    int gid = blockIdx.x * blockDim.x + threadIdx.x;
    int node = gid >> 5;
    if (node >= N_NODES) return;
    int lane = gid & 31;
    int b = batch[node];
    const float* xr = x + (size_t)node * DIM + lane * 4;
    const float* qr = q + (size_t)b * DIM + lane * 4;
    float s = xr[0] * qr[0] + xr[1] * qr[1] + xr[2] * qr[2] + xr[3] * qr[3];
    #pragma unroll
    for (int off = 16; off; off >>= 1) s += __shfl_xor(s, off, 32);
    if (lane == 0) e[node] = s;
}

// Per-graph softmax + weighted readout; writes q_star = [q, r].
__global__ __launch_bounds__(128) void att_kernel(const float* __restrict__ e,
                                                  const float* __restrict__ x,
                                                  const int* __restrict__ startv,
                                                  const int* __restrict__ endv,
                                                  const float* __restrict__ h1,
                                                  float* __restrict__ qstar) {
    int b = blockIdx.x, t = threadIdx.x;
    int s0 = startv[b], e0 = endv[b];
    __shared__ float red[DIM];
    // max
    float m = -3.4e38f;
    for (int n = s0 + t; n < e0; n += DIM) m = fmaxf(m, e[n]);
    red[t] = m; __syncthreads();
    for (int off = 64; off; off >>= 1) {
        if (t < off) red[t] = fmaxf(red[t], red[t + off]);
        __syncthreads();
    }
    m = red[0]; __syncthreads();
    // sum exp
    float ds = 0.0f;
    for (int n = s0 + t; n < e0; n += DIM) ds += expf(e[n] - m);
    red[t] = ds; __syncthreads();
    for (int off = 64; off; off >>= 1) {
        if (t < off) red[t] += red[t + off];
        __syncthreads();
    }
    float denom = red[0];
    float inv = (e0 > s0 && denom != 0.0f) ? 1.0f / denom : 0.0f;
    // weighted readout (threads coalesce across feature dim)
    float r = 0.0f;
    for (int n = s0; n < e0; ++n) {
        float a = expf(e[n] - m) * inv;
        r += a * x[(size_t)n * DIM + t];
    }
    qstar[(size_t)b * (2 * DIM) + t]       = h1[(size_t)b * DIM + t];
    qstar[(size_t)b * (2 * DIM) + DIM + t] = r;
}

// ---------------------------------------------------------------- launcher
static inline char* alignp(char* p) {
    return (char*)(((uintptr_t)p + 255) & ~(uintptr_t)255);
}

extern "C" void kernel_launch(void* const* d_in, const int* in_sizes, int n_in,
                              void* d_out, int out_size, void* d_ws, size_t ws_size,
                              hipStream_t stream) {
    const float* x_in   = (const float*)d_in[0];
    const int*   eidx   = (const int*)d_in[1];
    // d_in[2] = edge_attr (unused)
    const int*   batch  = (const int*)d_in[3];
    const float* w1_0   = (const float*)d_in[4];
    const float* b1_0   = (const float*)d_in[5];
    const float* w1     = (const float*)d_in[6];
    const float* b1     = (const float*)d_in[7];
    const float* w2     = (const float*)d_in[8];
    const float* b2     = (const float*)d_in[9];
    const float* gamma  = (const float*)d_in[10];
    const float* beta   = (const float*)d_in[11];
    const float* eps    = (const float*)d_in[12];
    const float* Wih0   = (const float*)d_in[13];
    const float* Whh0   = (const float*)d_in[14];
    const float* bih0   = (const float*)d_in[15];
    const float* bhh0   = (const float*)d_in[16];
    const float* Wih1   = (const float*)d_in[17];
    const float* Whh1   = (const float*)d_in[18];
    const float* bih1   = (const float*)d_in[19];
    const float* bhh1   = (const float*)d_in[20];

    const int* src = eidx;
    const int* dst = eidx + N_EDGES;

    const size_t featBytes = (size_t)N_NODES * DIM * sizeof(float);
    char* w = (char*)d_ws;
    float* xbuf  = (float*)alignp(w); w = (char*)xbuf  + featBytes;
    float* bufA  = (float*)alignp(w); w = (char*)bufA  + featBytes;
    float* bufB  = (float*)alignp(w); w = (char*)bufB  + featBytes;
    float* xmax  = (float*)alignp(w); w = (char*)xmax  + featBytes;
    float* w1pad = (float*)alignp(w); w = (char*)w1pad + DIM * DIM * sizeof(float);
    float* wf1   = (float*)alignp(w); w = (char*)wf1   + DIM * DIM * sizeof(float);
    float* wf2   = (float*)alignp(w); w = (char*)wf2   + DIM * DIM * sizeof(float);
    double* bnsum   = (double*)alignp(w); w = (char*)bnsum   + DIM * sizeof(double);
    double* bnsumsq = (double*)alignp(w); w = (char*)bnsumsq + DIM * sizeof(double);
    float* scale = (float*)alignp(w); w = (char*)scale + DIM * sizeof(float);
    float* shift = (float*)alignp(w); w = (char*)shift + DIM * sizeof(float);
    float* evec  = (float*)alignp(w); w = (char*)evec  + (size_t)N_NODES * sizeof(float);
    int* startv  = (int*)alignp(w);   w = (char*)startv + NUM_GRAPHS * sizeof(int);
    int* endv    = (int*)alignp(w);   w = (char*)endv   + NUM_GRAPHS * sizeof(int);
    // s2s state laid out contiguously so one zero-fill covers it
    float* s2s   = (float*)alignp(w);
    float* h0 = s2s;
    float* c0 = h0 + NUM_GRAPHS * DIM;
    float* h1 = c0 + NUM_GRAPHS * DIM;
    float* c1 = h1 + NUM_GRAPHS * DIM;
    float* qstar = c1 + NUM_GRAPHS * DIM;            // [64 x 256]
    const int s2s_floats = NUM_GRAPHS * DIM * 4 + NUM_GRAPHS * 2 * DIM;

    const int total = N_NODES * DIM;                 // 6,400,000
    const int TB = 256;
    const int gTotal = (total + TB - 1) / TB;
    const int gW = (DIM * DIM + TB - 1) / TB;
    dim3 gemmGrid((N_NODES / 16 + 3) / 4, 2);        // (782, 2): 4 waves x 4 n-tiles

    // ---- prepare padded input features and layer-0 weight
    pad_x_kernel<<<gTotal, TB, 0, stream>>>(x_in, xbuf);
    pad_w_kernel<<<gW, TB, 0, stream>>>(w1_0, w1pad);

    // ---- GIN layers
    for (int i = 0; i < NUM_LAYERS; ++i) {
        const float* W1p = (i == 0) ? w1pad : (w1 + (size_t)(i - 1) * DIM * DIM);
        const float* b1p = (i == 0) ? b1_0  : (b1 + (size_t)(i - 1) * DIM);
        const float* W2p = w2 + (size_t)i * DIM * DIM;
        const float* b2p = b2 + (size_t)i * DIM;

        // weight fragment swizzles (independent of the scatter phase)
        wfrag_kernel<<<gW, TB, 0, stream>>>(W1p, wf1);
        wfrag_kernel<<<gW, TB, 0, stream>>>(W2p, wf2);

        scale_x_kernel<<<gTotal, TB, 0, stream>>>(xbuf, eps + i, bufA, total);
        edge_scatter_kernel<<<(N_EDGES * 32) / TB, TB, 0, stream>>>(xbuf, src, dst, bufA);
        wmma_gemm4_leaky_kernel<<<gemmGrid, 128, 0, stream>>>(bufA, wf1, b1p, bufB, N_NODES);
        wmma_gemm4_leaky_kernel<<<gemmGrid, 128, 0, stream>>>(bufB, wf2, b2p, bufA, N_NODES);

        fill_zero_kernel<<<2, 256, 0, stream>>>((float*)bnsum, 2 * DIM * 2); // 2x128 doubles
        bn_stats_kernel<<<200, 128, 0, stream>>>(bufA, bnsum, bnsumsq);
        bn_apply_kernel<<<1, 128, 0, stream>>>(bnsum, bnsumsq, gamma + (size_t)i * DIM,
                                               beta + (size_t)i * DIM, scale, shift);
        bn_norm_kernel<<<gTotal, TB, 0, stream>>>(bufA, scale, shift, xbuf, xmax,
                                                  (i == 0) ? 1 : 0, total);
    }

    // ---- Set2Set over JK-max features (xmax)
    fill_zero_kernel<<<(s2s_floats + TB - 1) / TB, TB, 0, stream>>>(s2s, s2s_floats);
    fill_zero_kernel<<<1, 128, 0, stream>>>((float*)startv, 2 * NUM_GRAPHS);
    bounds_kernel<<<(N_NODES + TB - 1) / TB, TB, 0, stream>>>(batch, startv, endv);

    for (int step = 0; step < PROC_STEPS; ++step) {
        lstm_kernel<<<NUM_GRAPHS, 128, 0, stream>>>(qstar, 2 * DIM, Wih0, Whh0, bih0, bhh0, h0, c0);
        lstm_kernel<<<NUM_GRAPHS, 128, 0, stream>>>(h0, DIM, Wih1, Whh1, bih1, bhh1, h1, c1);
        escore_kernel<<<(N_NODES * 32 + TB - 1) / TB, TB, 0, stream>>>(xmax, h1, batch, evec);
        att_kernel<<<NUM_GRAPHS, 128, 0, stream>>>(evec, xmax, startv, endv, h1, qstar);
    }

    hipMemcpyAsync(d_out, qstar, (size_t)NUM_GRAPHS * 2 * DIM * sizeof(float),
                   hipMemcpyDeviceToDevice, stream);
}